// RyeElman_76819785056931
// MI455X (gfx1250) — compile-verified
//
#include <hip/hip_runtime.h>
#include <hip/hip_bf16.h>

// MI455X / gfx1250, wave32. FP32 WMMA (V_WMMA_F32_16X16X4_F32) keeps reference
// precision; problem is HBM-bound (~220MB @ 23.3TB/s ~ 9.5us floor), so the
// goal is streaming activations once while weights stay L2/register resident.
// Activation tile staging now uses the CDNA5 async global->LDS path
// (GLOBAL_LOAD_ASYNC_TO_LDS_B128, ASYNCcnt) to skip the VGPR round-trip.

typedef __attribute__((ext_vector_type(2))) float v2f;
typedef __attribute__((ext_vector_type(8))) float v8f;

#define ROWS    32      // batch rows per workgroup (2 M-tiles of 16)
#define MT      2
#define THREADS 288     // 9 waves: 8 invariant N-slices + 1 damp/equivariant
#define EPS     1e-6f

// LDS strides: INV_STRIDE ≡ 4 (mod 64), EQ_STRIDE ≡ -4 (mod 64) so that the
// WMMA A-fragment b64 reads (lanes 0-15: k,k+1; lanes 16-31: k+2,k+3) hit all
// 64 banks exactly once.
#define INV_STRIDE 196  // 192 cols + 4 pad
#define EQ_STRIDE  60   // 3 * 20 (17 channels padded to 20 for K%4==0)

__device__ __forceinline__ void async_copy16_to_lds(void* lds_dst, const void* gsrc) {
    // GLOBAL_LOAD_ASYNC_TO_LDS_B128: per-lane copy of 16B, global -> LDS,
    // tracked with ASYNCcnt (no VGPR data round-trip).
    unsigned lds_off = (unsigned)(unsigned long long)lds_dst; // low 32b = LDS offset
    asm volatile("global_load_async_to_lds_b128 %0, %1, off"
                 :
                 : "v"(lds_off), "v"(gsrc)
                 : "memory");
}

__device__ __forceinline__ void wait_async_lds() {
    asm volatile("s_wait_asynccnt 0x0" ::: "memory");
}

__global__ __launch_bounds__(THREADS)
void rye_elman_fused(const float* __restrict__ invIn,   // [B,64]
                     const float* __restrict__ eqIn,    // [B,3]
                     const float* __restrict__ invHid,  // [B,128]
                     const float* __restrict__ eqHid,   // [B,3,16]
                     const float* __restrict__ Wl,      // [17,128]
                     const float* __restrict__ Wr,      // [17,128]
                     const float* __restrict__ Wii,     // [192,128]
                     const float* __restrict__ bii,     // [128]
                     const float* __restrict__ Wee,     // [17,16]
                     const float* __restrict__ Wdamp,   // [192,16]
                     float* __restrict__ outInv,        // [B,128]
                     float* __restrict__ outEq,         // [B,3,16]
                     int B)
{
    __shared__ float sInv[ROWS * INV_STRIDE];
    __shared__ float sEq [ROWS * EQ_STRIDE];
    __shared__ float sXn [ROWS * EQ_STRIDE];

    const int tid  = threadIdx.x;
    const int row0 = blockIdx.x * ROWS;

    // ---- Stage inv_comb = [invIn | invHid] as [ROWS][192](+pad) ----
    // Direct async global->LDS B128 copies (both sides 16B aligned:
    // row base m*196*4 % 16 == 0, chunk step 16B).
    for (int i = tid; i < ROWS * 48; i += THREADS) {
        int m = i / 48, q = i % 48;
        int r = row0 + m; r = (r < B) ? r : (B - 1);
        const float4* src = (q < 16) ? ((const float4*)invIn ) + ((size_t)r * 16 + q)
                                     : ((const float4*)invHid) + ((size_t)r * 32 + (q - 16));
        async_copy16_to_lds(&sInv[m * INV_STRIDE + q * 4], src);
    }
    // ---- Stage eq_comb = [eqIn | eqHid] as [ROWS][3][20], pads zeroed ----
    // (interleave + zero-fill + 4B-aligned dst -> scalar path; overlaps with
    // the in-flight async inv-tile copies above)
    for (int i = tid; i < ROWS * EQ_STRIDE; i += THREADS) {
        int m = i / EQ_STRIDE, rr = i % EQ_STRIDE;
        int s = rr / 20, j = rr % 20;
        int r = row0 + m; r = (r < B) ? r : (B - 1);
        float v = 0.f;
        if (j == 0)      v = eqIn [(size_t)r * 3 + s];
        else if (j < 17) v = eqHid[((size_t)r * 3 + s) * 16 + (j - 1)];
        sEq[i] = v;
    }
    wait_async_lds();
    __syncthreads();

    // ---- xn = eq_comb / (sum_s eq^2 + eps), per (row, channel) ----
    for (int i = tid; i < ROWS * 20; i += THREADS) {
        int m = i / 20, j = i % 20;
        float e0 = sEq[m * EQ_STRIDE + 0 * 20 + j];
        float e1 = sEq[m * EQ_STRIDE + 1 * 20 + j];
        float e2 = sEq[m * EQ_STRIDE + 2 * 20 + j];
        float inv_n = (j < 17) ? (1.f / (e0 * e0 + e1 * e1 + e2 * e2 + EPS)) : 0.f;
        sXn[m * EQ_STRIDE + 0 * 20 + j] = e0 * inv_n;
        sXn[m * EQ_STRIDE + 1 * 20 + j] = e1 * inv_n;
        sXn[m * EQ_STRIDE + 2 * 20 + j] = e2 * inv_n;
    }
    __syncthreads();

    const int lane  = tid & 31;
    const int wave  = tid >> 5;
    const int half  = lane >> 4;      // half-wave select
    const int mrow  = lane & 15;      // A-fragment row (M)
    const int nlane = lane & 15;      // B/C column (N)
    const int koff  = half * 2;       // K sub-offset within a 4-wide step

    if (wave < 8) {
        // ================= invariant path: one 16-col slice of H ============
        const int n0 = wave * 16;

        // Register-resident B fragments (weights are L2-hot, loaded once/WG)
        v2f wii[48], wl[5], wr[5];
        #pragma unroll
        for (int kk = 0; kk < 48; ++kk) {
            int k = kk * 4 + koff;
            wii[kk].x = Wii[(k    ) * 128 + n0 + nlane];
            wii[kk].y = Wii[(k + 1) * 128 + n0 + nlane];
        }
        #pragma unroll
        for (int kk = 0; kk < 5; ++kk) {
            int k = kk * 4 + koff;
            wl[kk].x = (k     < 17) ? Wl[(k    ) * 128 + n0 + nlane] : 0.f;
            wl[kk].y = (k + 1 < 17) ? Wl[(k + 1) * 128 + n0 + nlane] : 0.f;
            wr[kk].x = (k     < 17) ? Wr[(k    ) * 128 + n0 + nlane] : 0.f;
            wr[kk].y = (k + 1 < 17) ? Wr[(k + 1) * 128 + n0 + nlane] : 0.f;
        }
        const float bias = bii[n0 + nlane];

        for (int mt = 0; mt < MT; ++mt) {
            const float* aBase = &sInv[(mt * 16 + mrow) * INV_STRIDE + koff];
            v8f acc = {};
            #pragma unroll
            for (int kk = 0; kk < 48; ++kk) {
                v2f a = *(const v2f*)(aBase + kk * 4);
                acc = __builtin_amdgcn_wmma_f32_16x16x4_f32(
                    false, a, false, wii[kk], (short)0, acc, false, false);
            }
            // projection: sum_s (xn@Wl) ⊙ (xn@Wr) — same C-tile layout
            v8f pacc = {};
            #pragma unroll
            for (int s = 0; s < 3; ++s) {
                const float* xBase = &sXn[(mt * 16 + mrow) * EQ_STRIDE + s * 20 + koff];
                v8f L = {}, R = {};
                #pragma unroll
                for (int kk = 0; kk < 5; ++kk) {
                    v2f a = *(const v2f*)(xBase + kk * 4);
                    L = __builtin_amdgcn_wmma_f32_16x16x4_f32(
                        false, a, false, wl[kk], (short)0, L, false, false);
                    R = __builtin_amdgcn_wmma_f32_16x16x4_f32(
                        false, a, false, wr[kk], (short)0, R, false, false);
                }
                #pragma unroll
                for (int e = 0; e < 8; ++e) pacc[e] += L[e] * R[e];
            }
            // epilogue: tanh(Cii + proj + b), C layout: VGPR e -> M = e + 8*half
            #pragma unroll
            for (int e = 0; e < 8; ++e) {
                int r = row0 + mt * 16 + e + half * 8;
                if (r < B)
                    outInv[(size_t)r * 128 + n0 + nlane] =
                        tanhf(acc[e] + pacc[e] + bias);
            }
        }
    } else {
        // ============ damp gate GEMM + fused equivariant update =============
        v2f wd[48], we[5];
        #pragma unroll
        for (int kk = 0; kk < 48; ++kk) {
            int k = kk * 4 + koff;
            wd[kk].x = Wdamp[(k    ) * 16 + nlane];
            wd[kk].y = Wdamp[(k + 1) * 16 + nlane];
        }
        #pragma unroll
        for (int kk = 0; kk < 5; ++kk) {
            int k = kk * 4 + koff;
            we[kk].x = (k     < 17) ? Wee[(k    ) * 16 + nlane] : 0.f;
            we[kk].y = (k + 1 < 17) ? Wee[(k + 1) * 16 + nlane] : 0.f;
        }

        for (int mt = 0; mt < MT; ++mt) {
            const float* aBase = &sInv[(mt * 16 + mrow) * INV_STRIDE + koff];
            v8f g = {};   // gate tile [16 rows][16 channels]
            #pragma unroll
            for (int kk = 0; kk < 48; ++kk) {
                v2f a = *(const v2f*)(aBase + kk * 4);
                g = __builtin_amdgcn_wmma_f32_16x16x4_f32(
                    false, a, false, wd[kk], (short)0, g, false, false);
            }
            #pragma unroll
            for (int s = 0; s < 3; ++s) {
                const float* eBase = &sEq[(mt * 16 + mrow) * EQ_STRIDE + s * 20 + koff];
                v8f T = {};  // eq_comb_s @ W_ee, same [row][channel] layout as g
                #pragma unroll
                for (int kk = 0; kk < 5; ++kk) {
                    v2f a = *(const v2f*)(eBase + kk * 4);
                    T = __builtin_amdgcn_wmma_f32_16x16x4_f32(
                        false, a, false, we[kk], (short)0, T, false, false);
                }
                #pragma unroll
                for (int e = 0; e < 8; ++e) {
                    int m = e + half * 8;
                    int r = row0 + mt * 16 + m;
                    float eqh = sEq[(mt * 16 + m) * EQ_STRIDE + s * 20 + 1 + nlane];
                    if (r < B)
                        outEq[((size_t)r * 3 + s) * 16 + nlane] = T[e] + eqh * g[e];
                }
            }
        }
    }
}

extern "C" void kernel_launch(void* const* d_in, const int* in_sizes, int n_in,
                              void* d_out, int out_size, void* d_ws, size_t ws_size,
                              hipStream_t stream) {
    (void)n_in; (void)out_size; (void)d_ws; (void)ws_size;
    const float* invIn  = (const float*)d_in[0];
    const float* eqIn   = (const float*)d_in[1];
    const float* invHid = (const float*)d_in[2];
    const float* eqHid  = (const float*)d_in[3];
    const float* Wl     = (const float*)d_in[4];
    const float* Wr     = (const float*)d_in[5];
    const float* Wii    = (const float*)d_in[6];
    const float* bii    = (const float*)d_in[7];
    const float* Wee    = (const float*)d_in[8];
    const float* Wdamp  = (const float*)d_in[9];

    const int B = in_sizes[0] / 64;          // invariant_input is [B, 64]
    float* outInv = (float*)d_out;           // [B,128]
    float* outEq  = outInv + (size_t)B * 128; // [B,3,16]

    const int grid = (B + ROWS - 1) / ROWS;
    hipLaunchKernelGGL(rye_elman_fused, dim3(grid), dim3(THREADS), 0, stream,
                       invIn, eqIn, invHid, eqHid, Wl, Wr, Wii, bii, Wee, Wdamp,
                       outInv, outEq, B);
}